// Conv2D_71485435674732
// MI455X (gfx1250) — compile-verified
//
#include <hip/hip_runtime.h>

// CDNA5 / gfx1250 implicit-GEMM conv2d 3x3 s1 p1, fp32 via V_WMMA_F32_16X16X4_F32
// GEMM view: M=Cout=128, N=NHW=131072, K=C*9=576.
// Double-buffered LDS staging, pixel-major B tile (contiguous k-pairs -> ds_load_b64),
// row stride padded to 20 floats to kill 4-way LDS bank conflicts.

typedef __attribute__((ext_vector_type(2))) float v2f;
typedef __attribute__((ext_vector_type(4))) float v4f;
typedef __attribute__((ext_vector_type(8))) float v8f;

#define C_IN  64
#define HH    64
#define WW    64
#define K_OUT 128
#define KRED  (C_IN * 9)   // 576
#define KSTEP 16
#define LDSA  20           // padded row stride (floats): conflict-free, 16B-aligned rows
#define LDSB  20

__global__ __launch_bounds__(256, 2)
void conv2d_wmma_f32(const float* __restrict__ x,
                     const float* __restrict__ wgt,
                     const float* __restrict__ bias,
                     float* __restrict__ out) {
    // A = weight tile [128 Cout][16 k], B = patch tile [64 w][16 k] (pixel-major)
    __shared__ float A_s[2][K_OUT * LDSA];   // 2 x 10 KB
    __shared__ float B_s[2][WW * LDSB];      // 2 x 5 KB

    const int tid  = threadIdx.x;
    const int lane = tid & 31;            // wave32
    const int wave = tid >> 5;            // 0..7 -> Cout tile (16 rows each)
    const int blk  = blockIdx.x;          // n*64 + h
    const int n    = blk >> 6;
    const int h    = blk & 63;

    const int m    = lane & 15;           // M/N position within 16
    const int khi  = (lane >> 4) << 1;    // 0 (lanes 0-15) or 2 (lanes 16-31), ISA A/B layout

    v8f acc0 = {}, acc1 = {}, acc2 = {}, acc3 = {};

    // Cooperative A load: thread -> 8 consecutive floats of weight row `ao`
    const int ao  = tid >> 1;             // 0..127
    const int akt = (tid & 1) << 3;       // 0 or 8
    // Cooperative B load: thread -> 4 consecutive pixels of k-row `br`
    const int br  = tid >> 4;             // 0..15
    const int bc  = (tid & 15) << 2;      // 0..60

    // ---- staging registers (filled one chunk ahead) ----
    v4f aw0, aw1;
    float bv0, bv1, bv2, bv3;

    auto load_regs = [&](int kkn) {
        // weights: contiguous 16-B-aligned (576*4 % 16 == 0, kkn%16==0, akt%4==0)
        const float* asrc = wgt + ao * KRED + kkn + akt;
        aw0 = *(const v4f*)(asrc);
        aw1 = *(const v4f*)(asrc + 4);
        // input patch row with zero padding
        const int k   = kkn + br;
        const int c   = k / 9;
        const int r9  = k - c * 9;
        const int kh  = r9 / 3;
        const int kw  = r9 - kh * 3;
        const int hin = h + kh - 1;
        bv0 = bv1 = bv2 = bv3 = 0.f;
        if ((unsigned)hin < (unsigned)HH) {
            const float* row = x + (((n * C_IN + c) * HH) + hin) * WW;
            const int w0 = bc + kw - 1;
            bv0 = ((unsigned)(w0)     < (unsigned)WW) ? row[w0]     : 0.f;
            bv1 = ((unsigned)(w0 + 1) < (unsigned)WW) ? row[w0 + 1] : 0.f;
            bv2 = ((unsigned)(w0 + 2) < (unsigned)WW) ? row[w0 + 2] : 0.f;
            bv3 = ((unsigned)(w0 + 3) < (unsigned)WW) ? row[w0 + 3] : 0.f;
        }
    };

    auto store_lds = [&](int buf) {
        float* Ab = &A_s[buf][0];
        float* Bb = &B_s[buf][0];
        *(v4f*)(&Ab[ao * LDSA + akt])     = aw0;   // 80B row stride -> 16B aligned
        *(v4f*)(&Ab[ao * LDSA + akt + 4]) = aw1;
        Bb[(bc + 0) * LDSB + br] = bv0;            // pixel-major scatter
        Bb[(bc + 1) * LDSB + br] = bv1;
        Bb[(bc + 2) * LDSB + br] = bv2;
        Bb[(bc + 3) * LDSB + br] = bv3;
    };

    // prologue: stage chunk 0
    load_regs(0);
    store_lds(0);

    int cur = 0;
    for (int kk = 0; kk < KRED; kk += KSTEP) {
        __syncthreads();                       // LDS buffer `cur` ready
        const int  nxt       = cur ^ 1;
        const bool have_next = (kk + KSTEP < KRED);

        if (have_next) load_regs(kk + KSTEP);  // VMEM issued, hidden under WMMAs
        if (kk + 2 * KSTEP < KRED)
            __builtin_prefetch(wgt + ao * KRED + kk + 2 * KSTEP + akt, 0, 1);

        const float* Ab = &A_s[cur][0];
        const float* Bb = &B_s[cur][0];
        #pragma unroll
        for (int kc = 0; kc < 4; ++kc) {
            const int kb = (kc << 2) + khi;
            // A fragment (16x4 f32): lane m = Cout row; VGPR0/1 = K kb/kb+1
            v2f a  = *(const v2f*)(&Ab[(wave * 16 + m) * LDSA + kb]);
            // B fragments (4x16 f32): lane m = pixel col; contiguous k-pair -> b64
            v2f b0 = *(const v2f*)(&Bb[( 0 + m) * LDSB + kb]);
            v2f b1 = *(const v2f*)(&Bb[(16 + m) * LDSB + kb]);
            v2f b2 = *(const v2f*)(&Bb[(32 + m) * LDSB + kb]);
            v2f b3 = *(const v2f*)(&Bb[(48 + m) * LDSB + kb]);
            acc0 = __builtin_amdgcn_wmma_f32_16x16x4_f32(false, a, false, b0, (short)0, acc0, false, false);
            acc1 = __builtin_amdgcn_wmma_f32_16x16x4_f32(false, a, false, b1, (short)0, acc1, false, false);
            acc2 = __builtin_amdgcn_wmma_f32_16x16x4_f32(false, a, false, b2, (short)0, acc2, false, false);
            acc3 = __builtin_amdgcn_wmma_f32_16x16x4_f32(false, a, false, b3, (short)0, acc3, false, false);
        }

        if (have_next) store_lds(nxt);         // different buffer: no race with readers of `cur`
        cur = nxt;
    }

    // Epilogue: C/D layout -> VGPR r holds M=r (lanes 0-15) or M=r+8 (lanes 16-31), N=lane%16
    const int mbase = wave * 16 + ((lane & 16) ? 8 : 0);
    #pragma unroll
    for (int r = 0; r < 8; ++r) {
        const int o    = mbase + r;
        const float bv = bias[o];
        float* orow = out + ((n * K_OUT + o) * HH + h) * WW;
        orow[ 0 + m] = acc0[r] + bv;
        orow[16 + m] = acc1[r] + bv;
        orow[32 + m] = acc2[r] + bv;
        orow[48 + m] = acc3[r] + bv;
    }
}

extern "C" void kernel_launch(void* const* d_in, const int* in_sizes, int n_in,
                              void* d_out, int out_size, void* d_ws, size_t ws_size,
                              hipStream_t stream) {
    const float* x    = (const float*)d_in[0];   // (32, 64, 64, 64)
    const float* wgt  = (const float*)d_in[1];   // (128, 64, 3, 3) == 128 x 576 row-major
    const float* bias = (const float*)d_in[2];   // (128,)
    float* out = (float*)d_out;                  // (32, 128, 64, 64)

    const int blocks = 32 * 64;                  // one (n, h) row per block
    conv2d_wmma_f32<<<dim3(blocks), dim3(256), 0, stream>>>(x, wgt, bias, out);
}